// MagnetLoss_50740743635423
// MI455X (gfx1250) — compile-verified
//
#include <hip/hip_runtime.h>
#include <math.h>

typedef __attribute__((ext_vector_type(2))) float v2f;
typedef __attribute__((ext_vector_type(8))) float v8f;

#define N_ROWS   16384
#define D_DIM    256
#define C_CLS    2048
#define MARGIN_F 1.0f
#define EPS_DIST 1e-5f
#define EPS_VAR  1e-4f
#define MAXM     16        // member-list capacity per class (N/C == 8 here)

// ---- workspace float layout (all regions fully written before read) ----
// f[0 .. N)                 row norms ||f_i||^2
// f[N .. 2N)                diag sqdist per row (for variance)
// f[2N .. 3N)               per-row loss
// f[3N .. 3N+C)             mean norms ||m_c||^2
// f[3N+C .. 3N+C+8)         scalars: [0]=cst, [1]=variance
// i[3N+C+8 .. +C*MAXM)      member row-indices per class (int)
// i[.. +C)                  member counts per class (int)
// f[MEANS ..]               16B-aligned class-means copy (WMMA B operand)
#define WS_ROWNORM  0
#define WS_DIAG     (N_ROWS)
#define WS_LOSSROW  (2 * N_ROWS)
#define WS_MEANNORM (3 * N_ROWS)
#define WS_SCALARS  (3 * N_ROWS + C_CLS)
#define WS_MEMBERS  (WS_SCALARS + 8)
#define WS_COUNTS   (WS_MEMBERS + C_CLS * MAXM)
#define WS_MEANS    (((WS_COUNTS + C_CLS) + 7) & ~7)   // 8-float (32B) aligned

// 1. One thread per class: scan labels in fixed i-order -> deterministic
//    member list. labels (64KB) is L2-resident; reads are wave-broadcast.
__global__ void build_members(const int* __restrict__ labels,
                              int* __restrict__ members,
                              int* __restrict__ counts) {
    const int c = blockIdx.x * blockDim.x + threadIdx.x;
    if (c >= C_CLS) return;
    int cnt = 0;
    for (int i = 0; i < N_ROWS; ++i) {
        if (labels[i] == c && cnt < MAXM) members[c * MAXM + cnt++] = i;
    }
    counts[c] = cnt;
}

// 2. One block per class: fixed-order gather -> mean, write aligned ws copy
//    + d_out copy, block-reduce ||m||^2.
__global__ void class_means(const float* __restrict__ feat,
                            const int* __restrict__ members,
                            const int* __restrict__ counts,
                            float* __restrict__ means_ws,
                            float* __restrict__ meannorm,
                            float* __restrict__ out_means) {
    const int c = blockIdx.x, t = threadIdx.x;
    const int cnt = counts[c];
    float s = 0.0f;
    for (int j = 0; j < cnt; ++j)
        s += feat[(size_t)members[c * MAXM + j] * D_DIM + t];
    const float v = (cnt > 0) ? s / (float)cnt : 0.0f;
    const size_t off = (size_t)c * D_DIM + t;
    means_ws[off]  = v;
    out_means[off] = v;
    float q = v * v;
#pragma unroll
    for (int o = 16; o > 0; o >>= 1) q += __shfl_down(q, o, 32);
    __shared__ float l8[8];
    if ((t & 31) == 0) l8[t >> 5] = q;
    __syncthreads();
    if (t == 0) {
        float tot = 0.0f;
#pragma unroll
        for (int w = 0; w < 8; ++w) tot += l8[w];
        meannorm[c] = tot;
    }
}

// 3. One block per row: ||f||^2 and diag sqdist ||f - m_label||^2 + eps.
__global__ void rownorm_diag(const float* __restrict__ feat,
                             const int* __restrict__ labels,
                             const float* __restrict__ means_ws,
                             float* __restrict__ rownorm,
                             float* __restrict__ diag) {
    const int i = blockIdx.x, t = threadIdx.x;
    const int lab = labels[i];
    const float f = feat[(size_t)i * D_DIM + t];
    const float m = means_ws[(size_t)lab * D_DIM + t];
    const float d = f - m;
    float sf = f * f, sd = d * d;
#pragma unroll
    for (int o = 16; o > 0; o >>= 1) {
        sf += __shfl_down(sf, o, 32);
        sd += __shfl_down(sd, o, 32);
    }
    __shared__ float lf[8], ldv[8];
    if ((t & 31) == 0) { lf[t >> 5] = sf; ldv[t >> 5] = sd; }
    __syncthreads();
    if (t == 0) {
        float a = 0.0f, b = 0.0f;
#pragma unroll
        for (int w = 0; w < 8; ++w) { a += lf[w]; b += ldv[w]; }
        rownorm[i] = a;
        diag[i]    = b + EPS_DIST;
    }
}

// 4. Single block, fixed-order strided sum + shuffle tree (deterministic):
//    variance and cst scalar.
__global__ void var_reduce(const float* __restrict__ diag,
                           float* __restrict__ scalars,
                           float* __restrict__ out_var) {
    const int t = threadIdx.x;
    float s = 0.0f;
    for (int i = t; i < N_ROWS; i += 256) s += diag[i];   // fixed order
#pragma unroll
    for (int o = 16; o > 0; o >>= 1) s += __shfl_down(s, o, 32);
    __shared__ float l8[8];
    if ((t & 31) == 0) l8[t >> 5] = s;
    __syncthreads();
    if (t == 0) {
        float tot = 0.0f;
#pragma unroll
        for (int w = 0; w < 8; ++w) tot += l8[w];
        const float variance = tot / (float)(N_ROWS - 1);
        scalars[0] = 1.0f / (-2.0f * (variance + EPS_VAR));
        scalars[1] = variance;
        *out_var   = variance;
    }
}

// 5. Fused fp32-WMMA GEMM + online softmax. Grid: N/16 WGs x 256 thr (8 waves).
//    Each wave processes 8 PAIRS of 16-class column tiles, sharing the A
//    fragment between the two tiles of a pair (1 ds b64 per 2 wmma).
__global__ void __launch_bounds__(256)
magnet_main(const float* __restrict__ feat,
            const int* __restrict__ labels,
            const float* __restrict__ means_ws,
            const float* __restrict__ meannorm,
            const float* __restrict__ rownorm,
            const float* __restrict__ scalars,
            float* __restrict__ loss_row) {
    __shared__ float sA[16 * D_DIM];        // 16 KB feature tile
    __shared__ float red_m[8][16];
    __shared__ float red_s[8][16];
    __shared__ float diag_lin[16];
    __shared__ int   labs[16];
    __shared__ float sqa_s[16];

    const int tid = threadIdx.x;
    const int r0  = blockIdx.x * 16;

    {   // cooperative 16x256 fp32 tile load (float4, coalesced)
        const float4* src = (const float4*)(feat + (size_t)r0 * D_DIM);
        float4* dst = (float4*)sA;
#pragma unroll
        for (int it = 0; it < 4; ++it) dst[tid + it * 256] = src[tid + it * 256];
    }
    if (tid < 16) {
        labs[tid]     = labels[r0 + tid];
        sqa_s[tid]    = rownorm[r0 + tid];
        diag_lin[tid] = 0.0f;
    }
    const float cst = scalars[0];           // uniform -> scalar load
    __syncthreads();

    const int wave = tid >> 5;
    const int lane = tid & 31;
    const int half = lane >> 4;
    const int l15  = lane & 15;

    float maxv[8], sums[8];
#pragma unroll
    for (int v = 0; v < 8; ++v) { maxv[v] = -INFINITY; sums[v] = 0.0f; }

#pragma unroll 1
    for (int it = 0; it < 8; ++it) {
        const int p  = it * 8 + wave;           // pair index [0,64)
        const int n0 = (2 * p) * 16 + l15;      // class column, tile 0 of pair
        const float* b0row = means_ws + (size_t)n0 * D_DIM;
        const float* b1row = b0row + 16 * D_DIM;   // n1 = n0 + 16
        if (it + 1 < 8)                          // global_prefetch next pair
            __builtin_prefetch(b0row + (size_t)256 * D_DIM, 0, 3);

        v8f acc[2] = {{}, {}};
        // A frag: row = l15, K = kb*4 + 2*half + {0,1}  (ISA 16x4 f32 layout)
        // B frags: col = l15, same K mapping; all 8B-aligned b64 loads.
#pragma unroll 4
        for (int kb = 0; kb < 64; ++kb) {
            const int k = kb * 4 + 2 * half;
            const v2f a  = *(const v2f*)(&sA[l15 * D_DIM + k]);
            const v2f b0 = *(const v2f*)(b0row + k);
            const v2f b1 = *(const v2f*)(b1row + k);
            acc[0] = __builtin_amdgcn_wmma_f32_16x16x4_f32(
                false, a, false, b0, (short)0, acc[0], false, false);
            acc[1] = __builtin_amdgcn_wmma_f32_16x16x4_f32(
                false, a, false, b1, (short)0, acc[1], false, false);
        }

        // epilogue for both tiles of the pair
#pragma unroll
        for (int j = 0; j < 2; ++j) {
            const int   n    = n0 + 16 * j;
            const float sq_b = meannorm[n];
#pragma unroll
            for (int v = 0; v < 8; ++v) {
                const int   m   = v + 8 * half;
                const float sqd = sqa_s[m] + sq_b - 2.0f * acc[j][v] + EPS_DIST;
                const bool  dia = (labs[m] == n);
                const float lin = cst * sqd - (dia ? MARGIN_F : 0.0f);
                if (dia) diag_lin[m] = lin;          // unique writer per row
                const float nm = fmaxf(maxv[v], lin);
                sums[v] = sums[v] * expf(maxv[v] - nm) + expf(lin - nm);
                maxv[v] = nm;
            }
        }
    }

    // combine the 16 lanes (different classes, same rows) in each half-wave
#pragma unroll
    for (int v = 0; v < 8; ++v) {
        float m = maxv[v], s = sums[v];
#pragma unroll
        for (int off = 1; off < 16; off <<= 1) {
            const float om = __shfl_xor(m, off, 32);
            const float os = __shfl_xor(s, off, 32);
            const float nm = fmaxf(m, om);
            s = s * expf(m - nm) + os * expf(om - nm);
            m = nm;
        }
        maxv[v] = m; sums[v] = s;
    }
    if (l15 == 0) {                              // lanes 0 (rows 0-7), 16 (rows 8-15)
#pragma unroll
        for (int v = 0; v < 8; ++v) {
            red_m[wave][v + 8 * half] = maxv[v];
            red_s[wave][v + 8 * half] = sums[v];
        }
    }
    __syncthreads();

    if (tid < 16) {                              // finish one row per thread
        const int row = tid;
        float M = -INFINITY;
#pragma unroll
        for (int w = 0; w < 8; ++w) M = fmaxf(M, red_m[w][row]);
        float S = 0.0f;
#pragma unroll
        for (int w = 0; w < 8; ++w) S += red_s[w][row] * expf(red_m[w][row] - M);
        const float a = expf(diag_lin[row] - M);
        const float b = S - a;                   // S counted all classes incl. diag
        float loss = -logf(EPS_VAR + a / (EPS_VAR + b));
        loss_row[r0 + row] = loss > 0.0f ? loss : 0.0f;   // hinge
    }
}

// 6. Single block, deterministic mean of per-row losses.
__global__ void loss_reduce(const float* __restrict__ loss_row,
                            float* __restrict__ out_loss) {
    const int t = threadIdx.x;
    float s = 0.0f;
    for (int i = t; i < N_ROWS; i += 256) s += loss_row[i];   // fixed order
#pragma unroll
    for (int o = 16; o > 0; o >>= 1) s += __shfl_down(s, o, 32);
    __shared__ float l8[8];
    if ((t & 31) == 0) l8[t >> 5] = s;
    __syncthreads();
    if (t == 0) {
        float tot = 0.0f;
#pragma unroll
        for (int w = 0; w < 8; ++w) tot += l8[w];
        *out_loss = tot / (float)N_ROWS;
    }
}

extern "C" void kernel_launch(void* const* d_in, const int* in_sizes, int n_in,
                              void* d_out, int out_size, void* d_ws, size_t ws_size,
                              hipStream_t stream) {
    (void)in_sizes; (void)n_in; (void)out_size; (void)ws_size;
    const float* feat   = (const float*)d_in[0];
    const int*   labels = (const int*)d_in[1];
    // d_in[2] = unique_labels = arange(C): identity mapping, not needed.

    float* out       = (float*)d_out;
    float* out_loss  = out;                        // [0]
    float* out_means = out + 1;                    // [1 .. 1+C*D)
    float* out_var   = out + 1 + (size_t)C_CLS * D_DIM;

    float* ws_f     = (float*)d_ws;
    float* rownorm  = ws_f + WS_ROWNORM;
    float* diag     = ws_f + WS_DIAG;
    float* loss_row = ws_f + WS_LOSSROW;
    float* meannorm = ws_f + WS_MEANNORM;
    float* scalars  = ws_f + WS_SCALARS;
    int*   members  = (int*)(ws_f + WS_MEMBERS);
    int*   counts   = (int*)(ws_f + WS_COUNTS);
    float* means_ws = ws_f + WS_MEANS;             // 32B-aligned means copy

    // 1. deterministic member lists per class
    build_members<<<C_CLS / 256, 256, 0, stream>>>(labels, members, counts);

    // 2. class means (fixed-order gather) + mean norms
    class_means<<<C_CLS, 256, 0, stream>>>(feat, members, counts,
                                           means_ws, meannorm, out_means);

    // 3. row norms + diagonal sqdist per row
    rownorm_diag<<<N_ROWS, 256, 0, stream>>>(feat, labels, means_ws,
                                             rownorm, diag);

    // 4. variance -> cst (deterministic tree)
    var_reduce<<<1, 256, 0, stream>>>(diag, scalars, out_var);

    // 5. fused WMMA GEMM + online-softmax loss (per-row losses)
    magnet_main<<<N_ROWS / 16, 256, 0, stream>>>(feat, labels, means_ws,
                                                 meannorm, rownorm, scalars,
                                                 loss_row);

    // 6. mean loss (deterministic tree)
    loss_reduce<<<1, 256, 0, stream>>>(loss_row, out_loss);
}